// GPCALayer_16965120820031
// MI455X (gfx1250) — compile-verified
//
#include <hip/hip_runtime.h>

typedef __attribute__((ext_vector_type(2))) float v2f;
typedef __attribute__((ext_vector_type(8))) float v8f;

#define NF 128   // NIN == NOUT == 128

// ---------------------------------------------------------------------------
// 1) Column means of x (for centering): one block per feature column.
// ---------------------------------------------------------------------------
__global__ __launch_bounds__(256) void col_mean_kernel(const float* __restrict__ x,
                                                       float* __restrict__ mean, int n) {
  __shared__ float sbuf[256];
  const int c = blockIdx.x;
  float s = 0.f;
  for (int r = threadIdx.x; r < n; r += 256) s += x[(long long)r * NF + c];
  sbuf[threadIdx.x] = s;
  __syncthreads();
  for (int off = 128; off > 0; off >>= 1) {
    if (threadIdx.x < off) sbuf[threadIdx.x] += sbuf[threadIdx.x + off];
    __syncthreads();
  }
  if (threadIdx.x == 0) mean[c] = sbuf[0] / (float)n;
}

// ---------------------------------------------------------------------------
// 2) Init (float4-vectorized): carry = x - mean, acc = 0, deg = 1 (self loop).
//    One thread per 4 consecutive features (b128 loads/stores).
// ---------------------------------------------------------------------------
__global__ __launch_bounds__(256) void init_kernel(const float4* __restrict__ x4,
                                                   const float4* __restrict__ mean4,
                                                   float4* __restrict__ carry4,
                                                   float4* __restrict__ acc4,
                                                   float* __restrict__ deg, int n) {
  const long long idx = (long long)blockIdx.x * 256 + threadIdx.x;   // float4 index
  const long long total = (long long)n * (NF / 4);
  if (idx >= total) return;
  const int c4 = (int)(idx & (NF / 4 - 1));
  const float4 xv = x4[idx];
  const float4 mv = mean4[c4];
  float4 cv;
  cv.x = xv.x - mv.x; cv.y = xv.y - mv.y; cv.z = xv.z - mv.z; cv.w = xv.w - mv.w;
  carry4[idx] = cv;
  acc4[idx] = make_float4(0.f, 0.f, 0.f, 0.f);
  if (c4 == 0) deg[idx >> 5] = 1.0f;
}

// ---------------------------------------------------------------------------
// 3) Row degrees of (A+I): atomic count per edge (deg pre-seeded with 1.0).
// ---------------------------------------------------------------------------
__global__ __launch_bounds__(256) void deg_kernel(const long long* __restrict__ erow,
                                                  float* __restrict__ deg, int e) {
  const int i = blockIdx.x * 256 + threadIdx.x;
  if (i < e) atomicAdd(&deg[(int)erow[i]], 1.0f);
}

__global__ __launch_bounds__(256) void invdeg_kernel(float* __restrict__ deg, int n) {
  const int i = blockIdx.x * 256 + threadIdx.x;
  if (i < n) {
    const float d = deg[i];
    deg[i] = d > 0.f ? 1.0f / d : 0.f;
  }
}

// ---------------------------------------------------------------------------
// 4) Edge scatter-add: one wave32 per edge; edge id is forced wave-uniform
//    via readfirstlane so the indices live in SGPRs (scalar loads). Each lane
//    moves a float4 (global_load_b128) and issues 4 global_atomic_add_f32
//    into acc[row]. carry/acc (51 MB each) are L2-resident on MI455X
//    (192 MB global L2), so the gather and the atomics land in L2.
// ---------------------------------------------------------------------------
__global__ __launch_bounds__(256) void edge_spmm_kernel(const long long* __restrict__ erow,
                                                        const long long* __restrict__ ecol,
                                                        const float* __restrict__ carry,
                                                        float* __restrict__ acc, int e) {
  const int wid = __builtin_amdgcn_readfirstlane((int)(threadIdx.x >> 5)); // 0..7
  const long long edge = (long long)blockIdx.x * 8 + wid;                   // uniform
  if (edge >= e) return;
  const int lane = (int)(threadIdx.x & 31);
  const int r = (int)erow[edge];
  const int c = (int)ecol[edge];
  const float4* __restrict__ src = (const float4*)(carry + (long long)c * NF);
  const float4 v = src[lane];
  float* __restrict__ dst = acc + (long long)r * NF + lane * 4;
  atomicAdd(dst + 0, v.x);
  atomicAdd(dst + 1, v.y);
  atomicAdd(dst + 2, v.z);
  atomicAdd(dst + 3, v.w);
}

// ---------------------------------------------------------------------------
// 5) Combine (float4-vectorized, in place):
//    carry = c1*invdeg*(acc + carry) + c2*(x - mean), then acc = 0.
// ---------------------------------------------------------------------------
__global__ __launch_bounds__(256) void combine_kernel(const float4* __restrict__ x4,
                                                      const float4* __restrict__ mean4,
                                                      const float* __restrict__ invdeg,
                                                      float4* __restrict__ carry4,
                                                      float4* __restrict__ acc4,
                                                      float c1, float c2, int n) {
  const long long idx = (long long)blockIdx.x * 256 + threadIdx.x;   // float4 index
  if (idx >= (long long)n * (NF / 4)) return;
  const int r = (int)(idx >> 5);
  const int c4 = (int)(idx & (NF / 4 - 1));
  const float s = c1 * invdeg[r];
  const float4 av = acc4[idx];
  const float4 cv = carry4[idx];
  const float4 xv = x4[idx];
  const float4 mv = mean4[c4];
  float4 o;
  o.x = s * (av.x + cv.x) + c2 * (xv.x - mv.x);
  o.y = s * (av.y + cv.y) + c2 * (xv.y - mv.y);
  o.z = s * (av.z + cv.z) + c2 * (xv.z - mv.z);
  o.w = s * (av.w + cv.w) + c2 * (xv.w - mv.w);
  carry4[idx] = o;
  acc4[idx] = make_float4(0.f, 0.f, 0.f, 0.f);
}

// ---------------------------------------------------------------------------
// 6) Final GEMM via V_WMMA_F32_16X16X4_F32.
//    Block = 256 threads = 8 waves; block handles a 16-row output stripe,
//    wave w handles output columns [16w, 16w+16).  K loop steps by 4.
//    A (16x4 f32): lane%16 = M row; lane/16 selects K pair -> one b64 load.
//    B (4x16 f32): lane%16 = N col; lane/16 selects K pair -> two b32 loads.
//    C/D (16x16 f32): VGPR r, lane l -> M = r + 8*(l>>4), N = l&15.
//    EXEC stays all-1s through the WMMA; the tail guard is a single scalar
//    branch (uniform per block), so the common path is 8 unguarded stores.
// ---------------------------------------------------------------------------
__global__ __launch_bounds__(256) void gemm_wmma_kernel(const float* __restrict__ X,
                                                        const float* __restrict__ W,
                                                        const float* __restrict__ bias,
                                                        float* __restrict__ out, int n) {
  const int wave = threadIdx.x >> 5;            // 0..7 -> N tile
  const int lane = threadIdx.x & 31;
  const int mBase = blockIdx.x * 16;
  const int nBase = wave * 16;
  const int row = lane & 15;                    // A-matrix M row / D column
  const int half = lane >> 4;                   // 0 or 1
  const int kk = half * 2;                      // K offset within 4-step
  const int col = lane & 15;

  int arow = mBase + row;
  if (arow >= n) arow = n - 1;                  // clamp: keep EXEC all-1s
  const float* __restrict__ Arow = X + (long long)arow * NF;

  v8f acc = {};
#pragma unroll 4
  for (int k = 0; k < NF; k += 4) {
    v2f a;
    a.x = Arow[k + kk];
    a.y = Arow[k + kk + 1];
    v2f b;
    b.x = W[(k + kk) * NF + nBase + col];
    b.y = W[(k + kk + 1) * NF + nBase + col];
    // (neg_a, A, neg_b, B, c_mod, C, reuse_a, reuse_b)
    acc = __builtin_amdgcn_wmma_f32_16x16x4_f32(false, a, false, b,
                                                (short)0, acc, false, false);
  }

  const float bv = bias[nBase + col];
  float* __restrict__ outp = out + (long long)(mBase + half * 8) * NF + nBase + col;
  if (mBase + 16 <= n) {                        // uniform branch: full tile
#pragma unroll
    for (int r = 0; r < 8; ++r) outp[(long long)r * NF] = acc[r] + bv;
  } else {                                      // rare ragged tail
#pragma unroll
    for (int r = 0; r < 8; ++r) {
      const int m = mBase + r + half * 8;
      if (m < n) out[(long long)m * NF + nBase + col] = acc[r] + bv;
    }
  }
}

// ---------------------------------------------------------------------------
// Host-side launch
// ---------------------------------------------------------------------------
extern "C" void kernel_launch(void* const* d_in, const int* in_sizes, int n_in,
                              void* d_out, int out_size, void* d_ws, size_t ws_size,
                              hipStream_t stream) {
  const float* x = (const float*)d_in[0];
  const long long* ei = (const long long*)d_in[1];   // int64 edge_index [2,E]
  const float* w = (const float*)d_in[2];
  const float* bias = (const float*)d_in[3];

  const int n = in_sizes[0] / NF;     // 100000
  const int e = in_sizes[1] / 2;      // 3200000
  const long long* erow = ei;
  const long long* ecol = ei + e;

  // Workspace layout: mean(1KB) | deg(n f32) | carry(n*128 f32) | acc(n*128 f32)
  char* ws = (char*)d_ws;
  size_t off = 0;
  float* mean = (float*)(ws + off);  off += 1024;
  float* deg = (float*)(ws + off);   off += ((size_t)n * 4 + 255) & ~(size_t)255;
  float* carry = (float*)(ws + off); off += (size_t)n * NF * 4;
  float* acc = (float*)(ws + off);   off += (size_t)n * NF * 4;
  (void)ws_size; (void)n_in; (void)out_size;

  const float alpha = 1.0f;
  const float c1 = alpha / (1.0f + alpha);
  const float c2 = 1.0f / (1.0f + alpha);

  col_mean_kernel<<<NF, 256, 0, stream>>>(x, mean, n);

  const long long total4 = (long long)n * (NF / 4);
  const int blkElem4 = (int)((total4 + 255) / 256);
  init_kernel<<<blkElem4, 256, 0, stream>>>((const float4*)x, (const float4*)mean,
                                            (float4*)carry, (float4*)acc, deg, n);

  deg_kernel<<<(e + 255) / 256, 256, 0, stream>>>(erow, deg, e);
  invdeg_kernel<<<(n + 255) / 256, 256, 0, stream>>>(deg, n);

  const int blkEdge = (e + 7) / 8;   // 8 edges (waves) per 256-thread block
  for (int t = 0; t < 50; ++t) {
    edge_spmm_kernel<<<blkEdge, 256, 0, stream>>>(erow, ecol, carry, acc, e);
    combine_kernel<<<blkElem4, 256, 0, stream>>>((const float4*)x, (const float4*)mean,
                                                 deg, (float4*)carry, (float4*)acc,
                                                 c1, c2, n);
  }

  gemm_wmma_kernel<<<(n + 15) / 16, 256, 0, stream>>>(carry, w, bias, (float*)d_out, n);
}